// TFTacotron2_45973329936829
// MI455X (gfx1250) — compile-verified
//
#include <hip/hip_runtime.h>

typedef __attribute__((ext_vector_type(2))) float v2f;
typedef __attribute__((ext_vector_type(4))) float v4f;
typedef __attribute__((ext_vector_type(8))) float v8f;

#define BB 128
#define TT 1024
#define DM 512
#define DQ 1024
#define DA 128
#define FF 32
#define KW 31

static __device__ __forceinline__ v8f wmma4(v2f a, v2f b, v8f c) {
  return __builtin_amdgcn_wmma_f32_16x16x4_f32(false, a, false, b, (short)0, c,
                                               false, false);
}

// ---------------- Kernel 1: pq = query @ Wq  ([128,1024] x [1024,128]) ----------------
__global__ __launch_bounds__(32) void pq_kernel(const float* __restrict__ q,
                                                const float* __restrict__ Wq,
                                                float* __restrict__ pq) {
  const int lane = threadIdx.x;
  const int l  = lane & 15;
  const int hi = lane >> 4;
  const int row0 = blockIdx.x * 16;  // over B
  const int n0   = blockIdx.y * 16;  // over DA
  v8f acc = {};
  for (int k = 0; k < DQ; k += 8) {
    const int ka = k + 4 * hi;  // this lane-half's 4 physical k values
    v4f a = *(const v4f*)(q + (row0 + l) * DQ + ka);
    const float* wp = Wq + ka * DA + n0 + l;
    v2f b0 = {wp[0], wp[DA]};
    v2f b1 = {wp[2 * DA], wp[3 * DA]};
    acc = wmma4(a.xy, b0, acc);  // K-slots = {k,k+1 | k+4,k+5}
    acc = wmma4(a.zw, b1, acc);  // K-slots = {k+2,k+3 | k+6,k+7}
  }
#pragma unroll
  for (int r = 0; r < 8; ++r)
    pq[(row0 + r + 8 * hi) * DA + n0 + l] = acc[r];
}

// ---------------- Kernel 2: fused conv + loc + keys + energy + softmax + context ------
__global__ __launch_bounds__(256) void attn_kernel(
    const float* __restrict__ state, const float* __restrict__ memory,
    const float* __restrict__ Wmem, const float* __restrict__ conv_k,
    const float* __restrict__ Wloc, const float* __restrict__ v_w,
    const float* __restrict__ v_b, const float* __restrict__ pq,
    float* __restrict__ ctx_out, float* __restrict__ align_out,
    float* __restrict__ nstate_out) {
  __shared__ float s_state[TT + KW - 1];
  __shared__ float s_ck[KW * FF];
  __shared__ float s_wloc[FF * DA];
  __shared__ float s_vw[DA];
  __shared__ float s_pq[DA];
  __shared__ float s_e[TT];
  __shared__ float s_red[8];
  __shared__ v4f   s_ctx[DM / 4];  // 128 v4f = 2KB context combine buffer

  const int tid  = threadIdx.x;
  const int b    = blockIdx.x;
  const int wave = tid >> 5;
  const int lane = tid & 31;
  const int l    = lane & 15;
  const int hi   = lane >> 4;

  for (int i = tid; i < TT; i += 256) s_state[i + 15] = state[b * TT + i];
  if (tid < 15) { s_state[tid] = 0.f; s_state[TT + 15 + tid] = 0.f; }
  for (int i = tid; i < KW * FF; i += 256) s_ck[i] = conv_k[i];
  for (int i = tid; i < FF * DA; i += 256) s_wloc[i] = Wloc[i];
  if (tid < DA) { s_vw[tid] = v_w[tid]; s_pq[tid] = pq[b * DA + tid]; }
  __syncthreads();

  const float vb = v_b[0];
  const float* __restrict__ memb = memory + (size_t)b * TT * DM;

  // ---- energy: 4 passes, each wave handles rows [t0, t0+32) ----
  for (int p = 0; p < 4; ++p) {
    const int t0 = (p * 8 + wave) * 32;
    v8f c0[8], c1[8];
#pragma unroll
    for (int i = 0; i < 8; ++i) { c0[i] = (v8f){}; c1[i] = (v8f){}; }

    if (p < 3)  // prefetch next pass's A rows toward L2/WGP$
      __builtin_prefetch(memb + (size_t)(t0 + 256 + l) * DM + 4 * hi, 0, 1);

    // keys: memory[b, t0:t0+32, :] @ Wmem  (K = 512, double-k-steps of 8)
    const float* arow0 = memb + (size_t)(t0 + l) * DM;
    const float* arow1 = memb + (size_t)(t0 + 16 + l) * DM;
    for (int k = 0; k < DM; k += 8) {
      const int ka = k + 4 * hi;
      v4f a0 = *(const v4f*)(arow0 + ka);
      v4f a1 = *(const v4f*)(arow1 + ka);
#pragma unroll
      for (int i = 0; i < 8; ++i) {
        const float* wp = Wmem + ka * DA + i * 16 + l;
        v2f b0 = {wp[0], wp[DA]};
        v2f b1 = {wp[2 * DA], wp[3 * DA]};
        c0[i] = wmma4(a0.xy, b0, c0[i]);
        c1[i] = wmma4(a1.xy, b0, c1[i]);
        c0[i] = wmma4(a0.zw, b1, c0[i]);
        c1[i] = wmma4(a1.zw, b1, c1[i]);
      }
    }

    // conv features: this lane's 16 channels (group g: channels 8g+4*hi .. +3)
    v4f f0[4], f1[4];
#pragma unroll
    for (int g = 0; g < 4; ++g) { f0[g] = (v4f){}; f1[g] = (v4f){}; }
    {
      const float* sp0 = s_state + (t0 + l);  // sp[j] = state[t + j - 15] (padded)
      const float* sp1 = sp0 + 16;
#pragma unroll
      for (int j = 0; j < KW; ++j) {
        const float s0 = sp0[j];
        const float s1 = sp1[j];
#pragma unroll
        for (int g = 0; g < 4; ++g) {
          v4f ckv = *(const v4f*)(s_ck + j * FF + 8 * g + 4 * hi);
          f0[g] += s0 * ckv;
          f1[g] += s1 * ckv;
        }
      }
    }
    // loc: f @ Wloc accumulated into the same tiles (K = 32)
#pragma unroll
    for (int g = 0; g < 4; ++g) {
      const int ka = 8 * g + 4 * hi;
#pragma unroll
      for (int i = 0; i < 8; ++i) {
        const float* wp = s_wloc + ka * DA + i * 16 + l;
        v2f b0 = {wp[0], wp[DA]};
        v2f b1 = {wp[2 * DA], wp[3 * DA]};
        c0[i] = wmma4(f0[g].xy, b0, c0[i]);
        c1[i] = wmma4(f1[g].xy, b0, c1[i]);
        c0[i] = wmma4(f0[g].zw, b1, c0[i]);
        c1[i] = wmma4(f1[g].zw, b1, c1[i]);
      }
    }

    // energy rows: tanh(keys+loc+pq) . v_w (+ v_b)
#pragma unroll
    for (int r = 0; r < 8; ++r) {
      float p0 = 0.f, p1 = 0.f;
#pragma unroll
      for (int i = 0; i < 8; ++i) {
        const float pqv = s_pq[i * 16 + l];
        const float vwv = s_vw[i * 16 + l];
        p0 = fmaf(tanhf(c0[i][r] + pqv), vwv, p0);
        p1 = fmaf(tanhf(c1[i][r] + pqv), vwv, p1);
      }
      p0 += __shfl_xor(p0, 1, 32); p1 += __shfl_xor(p1, 1, 32);
      p0 += __shfl_xor(p0, 2, 32); p1 += __shfl_xor(p1, 2, 32);
      p0 += __shfl_xor(p0, 4, 32); p1 += __shfl_xor(p1, 4, 32);
      p0 += __shfl_xor(p0, 8, 32); p1 += __shfl_xor(p1, 8, 32);
      if (l == 0) {
        s_e[t0 + r + 8 * hi]      = p0 + vb;
        s_e[t0 + 16 + r + 8 * hi] = p1 + vb;
      }
    }
  }
  __syncthreads();

  // ---- softmax over T: each thread owns 4 contiguous elements ----
  const int i4 = tid * 4;
  v4f ev = *(const v4f*)(s_e + i4);
  float lmax = fmaxf(fmaxf(ev.x, ev.y), fmaxf(ev.z, ev.w));
#pragma unroll
  for (int off = 16; off >= 1; off >>= 1) lmax = fmaxf(lmax, __shfl_xor(lmax, off, 32));
  if (lane == 0) s_red[wave] = lmax;
  __syncthreads();
  float gmax = s_red[0];
#pragma unroll
  for (int w = 1; w < 8; ++w) gmax = fmaxf(gmax, s_red[w]);

  v4f ex;
  ex.x = __expf(ev.x - gmax); ex.y = __expf(ev.y - gmax);
  ex.z = __expf(ev.z - gmax); ex.w = __expf(ev.w - gmax);
  float lsum = (ex.x + ex.y) + (ex.z + ex.w);
#pragma unroll
  for (int off = 16; off >= 1; off >>= 1) lsum += __shfl_xor(lsum, off, 32);
  __syncthreads();  // all gmax reads of s_red done
  if (lane == 0) s_red[wave] = lsum;
  __syncthreads();
  float gsum = 0.f;
#pragma unroll
  for (int w = 0; w < 8; ++w) gsum += s_red[w];
  const float inv = 1.f / gsum;

  // alignments + cumulate (b128 stores)
  {
    v4f a = ex * inv;
    *(v4f*)(s_e + i4) = a;
    *(v4f*)(align_out + b * TT + i4) = a;
    v4f st = *(const v4f*)(state + b * TT + i4);
    *(v4f*)(nstate_out + b * TT + i4) = a + st;
  }
  __syncthreads();

  // ---- context[b,:]: one b128 column-group per thread, T split in two halves ----
  {
    const int cg   = tid & 127;   // column group (DM/4 = 128)
    const int half = tid >> 7;
    const v4f* mem4 = (const v4f*)memb;
    v4f acc = {};
    const int tbeg = half * (TT / 2), tend = tbeg + (TT / 2);
    for (int t = tbeg; t < tend; ++t)
      acc += s_e[t] * mem4[t * (DM / 4) + cg];
    if (half == 1) s_ctx[cg] = acc;
    __syncthreads();
    if (half == 0)
      ((v4f*)(ctx_out + b * DM))[cg] = acc + s_ctx[cg];
  }
}

extern "C" void kernel_launch(void* const* d_in, const int* in_sizes, int n_in,
                              void* d_out, int out_size, void* d_ws, size_t ws_size,
                              hipStream_t stream) {
  (void)in_sizes; (void)n_in; (void)out_size; (void)ws_size;
  const float* query  = (const float*)d_in[0];
  const float* state  = (const float*)d_in[1];
  /* d_in[2] prev_max_alignments: unused by the reference computation */
  const float* memory = (const float*)d_in[3];
  const float* Wq     = (const float*)d_in[4];
  const float* Wmem   = (const float*)d_in[5];
  const float* conv_k = (const float*)d_in[6];
  const float* Wloc   = (const float*)d_in[7];
  const float* v_w    = (const float*)d_in[8];
  const float* v_b    = (const float*)d_in[9];

  float* out    = (float*)d_out;
  float* ctx    = out;                  // [B, DM]
  float* align  = out + BB * DM;        // [B, T]
  float* nstate = align + BB * TT;      // [B, T]
  float* pq     = (float*)d_ws;         // [B, DA] scratch

  pq_kernel<<<dim3(BB / 16, DA / 16), 32, 0, stream>>>(query, Wq, pq);
  attn_kernel<<<BB, 256, 0, stream>>>(state, memory, Wmem, conv_k, Wloc, v_w, v_b,
                                      pq, ctx, align, nstate);
}